// GAT_KDE_14766097563859
// MI455X (gfx1250) — compile-verified
//
#include <hip/hip_runtime.h>
#include <math.h>

// ---------------- static config (matches reference) ----------------
#define NODES   2048
#define NGRAPH  4
#define NPG     512
#define NEDGE   32768
#define EPN     (NEDGE + NODES)   // 34816 (edges + self loops)
#define DIM     128
#define HEADS   4
#define HID     32
#define OUTD    32
#define GRIDP   500
#define NQ      20
#define NEG_SLOPE 0.2f
#define INV_SQRT_2PI 0.3989422804014327f
#define N_POW_M02 0.2871746f      // 512^(-0.2)

typedef float v2f __attribute__((ext_vector_type(2)));
typedef float v8f __attribute__((ext_vector_type(8)));

// ---------------- f32 WMMA GEMM:  [2048,128] @ [128,128] -> [2048,128] ----
// Block = 8 waves covering one 16-row M tile; waves own the 8 N tiles.
// The shared 16x128 A tile is staged via CDNA5 async global->LDS copies
// (GLOBAL_LOAD_ASYNC_TO_LDS_B128, ASYNCcnt), then each wave runs 32
// V_WMMA_F32_16X16X4_F32 with A fragments from LDS and B from L2.
__global__ __launch_bounds__(256)
void gemm_wmma_f32(const float* __restrict__ A, const float* __restrict__ W,
                   float* __restrict__ O) {
  __shared__ __align__(16) float sA[16 * DIM];   // 8 KB A tile [16 x 128]
  const int tid  = threadIdx.x;
  const int wave = tid >> 5;
  const int lane = tid & 31;
  const int m0   = blockIdx.x * 16;             // 128 M tiles
  const int n0   = wave * 16;                   // 8 N tiles per block
  const int half = lane >> 4;                   // K-half select
  const int r    = lane & 15;                   // row (A) / col (B,C)

  // ---- async stage of the A tile: 2048 floats = 2x b128 per thread ----
  {
    const float* g0 = A + (size_t)m0 * DIM + tid * 4;
    const float* g1 = g0 + 1024;
    // LDS-aperture flat address carries the workgroup-relative LDS offset
    // in its low 32 bits (ISA 10.2), which is what dsaddr (VDST) wants.
    unsigned l0 = (unsigned)(size_t)(void*)&sA[tid * 4];
    unsigned l1 = (unsigned)(size_t)(void*)&sA[tid * 4 + 1024];
    asm volatile("global_load_async_to_lds_b128 %0, %1, off"
                 :: "v"(l0), "v"(g0) : "memory");
    asm volatile("global_load_async_to_lds_b128 %0, %1, off"
                 :: "v"(l1), "v"(g1) : "memory");
    asm volatile("s_wait_asynccnt 0x0" ::: "memory");
  }
  __syncthreads();

  v8f acc = {0.f, 0.f, 0.f, 0.f, 0.f, 0.f, 0.f, 0.f};
  const float* arow = &sA[r * DIM + 2 * half];             // LDS
  const float* bcol = W + (2 * half) * DIM + (n0 + r);     // global (L2)

#pragma unroll 8
  for (int k = 0; k < DIM; k += 4) {
    v2f a, b;
    a.x = arow[k];
    a.y = arow[k + 1];
    b.x = bcol[k * DIM];
    b.y = bcol[(k + 1) * DIM];
    // (neg_a, A, neg_b, B, c_mod, C, reuse_a, reuse_b)
    acc = __builtin_amdgcn_wmma_f32_16x16x4_f32(false, a, false, b,
                                                (short)0, acc, false, false);
  }
#pragma unroll
  for (int vr = 0; vr < 8; ++vr) {
    const int row = m0 + vr + 8 * half;         // C/D layout: M = vgpr + 8*half
    O[row * DIM + n0 + r] = acc[vr];
  }
}

// ---------------- attention logits: al_s/al_d [N,H] ----------------
__global__ __launch_bounds__(256)
void att_logits(const float* __restrict__ xh, const float* __restrict__ a_src,
                const float* __restrict__ a_dst, float* __restrict__ al_s,
                float* __restrict__ al_d) {
  int idx = blockIdx.x * blockDim.x + threadIdx.x;   // NODES*HEADS = 8192
  if (idx >= NODES * HEADS) return;
  const int n = idx >> 2, h = idx & 3;
  const float* v  = xh + n * DIM + h * HID;
  const float* as = a_src + h * HID;
  const float* ad = a_dst + h * HID;
  float ss = 0.f, sd = 0.f;
#pragma unroll 8
  for (int c = 0; c < HID; ++c) { float x = v[c]; ss += x * as[c]; sd += x * ad[c]; }
  al_s[idx] = ss;
  al_d[idx] = sd;
}

// ---------------- per-layer init: out=bias, e_max=-inf, denom=0 ----------
__global__ __launch_bounds__(256)
void init_layer(const float* __restrict__ bias, float* __restrict__ out,
                float* __restrict__ e_max, float* __restrict__ denom) {
  int idx = blockIdx.x * blockDim.x + threadIdx.x;   // NODES*DIM
  if (idx >= NODES * DIM) return;
  const int c = idx & (DIM - 1);
  out[idx] = bias[c];
  if (c < HEADS) {
    const int n = idx >> 7;
    e_max[n * HEADS + c] = -3.0e38f;
    denom[n * HEADS + c] = 0.f;
  }
}

// float atomic max via sign-aware integer atomics (always available)
__device__ inline void atomicMaxF(float* addr, float value) {
  if (value >= 0.f) atomicMax((int*)addr, __float_as_int(value));
  else              atomicMin((unsigned int*)addr, __float_as_uint(value));
}

__device__ inline void edge_sd(int i, const int* __restrict__ src,
                               const int* __restrict__ dst, int& s, int& d) {
  if (i < NEDGE) { s = src[i]; d = dst[i]; }
  else           { s = i - NEDGE; d = s; }        // self loops
}

// ---------------- edge pass 1: segment max (stable softmax) ----------
__global__ __launch_bounds__(256)
void edge_max_k(const int* __restrict__ src, const int* __restrict__ dst,
                const float* __restrict__ al_s, const float* __restrict__ al_d,
                float* __restrict__ e_max) {
  int idx = blockIdx.x * blockDim.x + threadIdx.x;   // EPN*HEADS
  if (idx >= EPN * HEADS) return;
  const int i = idx >> 2, h = idx & 3;
  int s, d; edge_sd(i, src, dst, s, d);
  float e = al_s[s * HEADS + h] + al_d[d * HEADS + h];
  e = e > 0.f ? e : NEG_SLOPE * e;
  atomicMaxF(&e_max[d * HEADS + h], e);
}

// ---------------- edge pass 2: exp + denom -------------------------
__global__ __launch_bounds__(256)
void edge_exp_k(const int* __restrict__ src, const int* __restrict__ dst,
                const float* __restrict__ al_s, const float* __restrict__ al_d,
                const float* __restrict__ e_max, float* __restrict__ ebuf,
                float* __restrict__ denom) {
  int idx = blockIdx.x * blockDim.x + threadIdx.x;   // EPN*HEADS
  if (idx >= EPN * HEADS) return;
  const int i = idx >> 2, h = idx & 3;
  int s, d; edge_sd(i, src, dst, s, d);
  float e = al_s[s * HEADS + h] + al_d[d * HEADS + h];
  e = e > 0.f ? e : NEG_SLOPE * e;
  const float ev = __expf(e - e_max[d * HEADS + h]);
  ebuf[idx] = ev;
  atomicAdd(&denom[d * HEADS + h], ev);
}

// ---------------- edge pass 3: alpha-weighted scatter ---------------
__global__ __launch_bounds__(256)
void edge_scatter_k(const int* __restrict__ src, const int* __restrict__ dst,
                    const float* __restrict__ ebuf, const float* __restrict__ denom,
                    const float* __restrict__ xh, float* __restrict__ out) {
  int idx = blockIdx.x * blockDim.x + threadIdx.x;   // EPN*DIM
  if (idx >= EPN * DIM) return;
  const int i = idx >> 7, c = idx & 127, h = c >> 5;
  int s, d; edge_sd(i, src, dst, s, d);
  const float alpha = ebuf[i * HEADS + h] / (denom[d * HEADS + h] + 1e-16f);
  atomicAdd(&out[d * DIM + c], alpha * xh[s * DIM + c]);
}

__global__ __launch_bounds__(256)
void relu_k(float* __restrict__ v) {
  int idx = blockIdx.x * blockDim.x + threadIdx.x;
  if (idx < NODES * DIM) v[idx] = fmaxf(v[idx], 0.f);
}

// ---- fused readout stats: mean/max pool + linear head + KDE stats ----
// One pass over [512,128] per graph produces sum/max/min/sumsq, emits the
// pooled linear head AND the KDE min/max/bandwidth arrays.
__global__ __launch_bounds__(128)
void readout_stats(const float* __restrict__ cur, const float* __restrict__ pool_w,
                   const float* __restrict__ lpW, const float* __restrict__ lpb,
                   float* __restrict__ hout, float* __restrict__ mnb,
                   float* __restrict__ mxb, float* __restrict__ hband) {
  __shared__ float swp[DIM];
  const int g = blockIdx.x, c = threadIdx.x;
  const float* base = cur + (size_t)g * NPG * DIM + c;
  float sum = 0.f, mx = -3.0e38f, mn = 3.0e38f, s2 = 0.f;
  for (int n = 0; n < NPG; ++n) {
    float v = base[n * DIM];
    sum += v; s2 += v * v;
    mx = fmaxf(mx, v); mn = fminf(mn, v);
  }
  const float mean = sum * (1.f / NPG);
  float var = s2 * (1.f / NPG) - mean * mean;
  var = var > 0.f ? var : 0.f;
  mnb[g * DIM + c]   = mn - 1e-6f;
  mxb[g * DIM + c]   = mx + 1e-6f;
  hband[g * DIM + c] = 1.06f * (sqrtf(var) + (1e-8f / 3.0f)) * N_POW_M02;
  swp[c] = pool_w[0] * mean + pool_w[1] * mx;
  __syncthreads();
  if (c < OUTD) {
    float acc = lpb[c];
    for (int k = 0; k < DIM; ++k) acc += swp[k] * lpW[k * OUTD + c];
    hout[g * OUTD + c] = acc;
  }
}

// ---------------- KDE main: density -> cdf -> soft quantiles --------
// block per (graph, dim) = 512 blocks; node column staged in LDS.
__global__ __launch_bounds__(256)
void kde_density(const float* __restrict__ cur, const float* __restrict__ mnb,
                 const float* __restrict__ mxb, const float* __restrict__ hband,
                 float* __restrict__ kf) {
  __shared__ float sx[NPG];
  __shared__ float sdens[GRIDP];
  __shared__ float sred[256];
  __shared__ float sred2[256];
  const int g = blockIdx.x >> 7;
  const int d = blockIdx.x & 127;
  const int t = threadIdx.x;

  const float* base = cur + (size_t)g * NPG * DIM + d;
  for (int n = t; n < NPG; n += 256) sx[n] = base[n * DIM];
  __syncthreads();

  const float mn = mnb[g * DIM + d];
  const float scale = mxb[g * DIM + d] - mn;
  const float h = hband[g * DIM + d];
  const float invh = 1.0f / h;
  const float norm = invh * INV_SQRT_2PI * (1.f / NPG);
  const float gstep = 1.0f / (GRIDP - 1);

  for (int gp = t; gp < GRIDP; gp += 256) {
    const float xg = mn + scale * ((float)gp * gstep);
    float acc = 0.f;
#pragma unroll 4
    for (int n = 0; n < NPG; ++n) {
      const float u = (xg - sx[n]) * invh;
      acc += __expf(-0.5f * u * u);
    }
    sdens[gp] = acc * norm;
  }
  __syncthreads();

  if (t == 0) {                       // cumsum over 500 grid points
    float run = 0.f;
    for (int gp = 0; gp < GRIDP; ++gp) { run += sdens[gp]; sdens[gp] = run; }
  }
  __syncthreads();
  const float invlast = 1.0f / fmaxf(sdens[GRIDP - 1], 1e-8f);

  for (int q = 0; q < NQ; ++q) {
    const float Qv = (float)q * (1.0f / (NQ - 1));
    float wsum = 0.f, gwsum = 0.f;
    for (int gp = t; gp < GRIDP; gp += 256) {
      const float cdf = sdens[gp] * invlast;
      const float dist = fabsf(cdf - Qv);
      const float w = 1.0f / (1.0f + __expf(100.0f * dist));   // sigmoid(-100*dist)
      const float xg = mn + scale * ((float)gp * gstep);
      wsum += w; gwsum += w * xg;
    }
    sred[t] = wsum; sred2[t] = gwsum;
    __syncthreads();
    for (int off = 128; off > 0; off >>= 1) {
      if (t < off) { sred[t] += sred[t + off]; sred2[t] += sred2[t + off]; }
      __syncthreads();
    }
    if (t == 0) kf[((size_t)g * DIM + d) * NQ + q] = sred2[0] / (sred[0] + 1e-8f);
    __syncthreads();
  }
}

// ---------------- KDE projector: [G,2560] @ [2560,32] --------------
__global__ __launch_bounds__(32)
void kde_proj(const float* __restrict__ kf, const float* __restrict__ kW,
              const float* __restrict__ kb, float* __restrict__ kout) {
  const int g = blockIdx.x, o = threadIdx.x;
  if (o >= OUTD) return;
  float acc = kb[o];
  const float* f = kf + (size_t)g * DIM * NQ;
  for (int j = 0; j < DIM * NQ; ++j) acc += f[j] * kW[j * OUTD + o];
  kout[g * OUTD + o] = acc;
}

// ---------------- final risk head ----------------------------------
__global__ __launch_bounds__(32)
void final_risk(const float* __restrict__ hr, const float* __restrict__ kr,
                const float* __restrict__ beta, const float* __restrict__ h0,
                float* __restrict__ out) {
  const int g = threadIdx.x;
  if (g >= NGRAPH) return;
  float acc = h0[0];
  for (int o = 0; o < OUTD; ++o) {
    const int i = g * OUTD + o;
    const float m = (hr[i] + hr[NGRAPH * OUTD + i] + hr[2 * NGRAPH * OUTD + i]) * (1.f / 3.f);
    const float k = (kr[i] + kr[NGRAPH * OUTD + i] + kr[2 * NGRAPH * OUTD + i]) * (1.f / 3.f);
    acc += (m + k) * beta[o];
  }
  out[g] = acc;
}

// ---------------- host orchestration --------------------------------
extern "C" void kernel_launch(void* const* d_in, const int* in_sizes, int n_in,
                              void* d_out, int out_size, void* d_ws, size_t ws_size,
                              hipStream_t stream) {
  const float* x      = (const float*)d_in[0];
  const int*   eidx   = (const int*)d_in[1];     // [2,E]; src = eidx, dst = eidx+E
  const float* W0     = (const float*)d_in[3];
  const float* as0    = (const float*)d_in[4];
  const float* ad0    = (const float*)d_in[5];
  const float* b0     = (const float*)d_in[6];
  const float* W1     = (const float*)d_in[7];
  const float* as1    = (const float*)d_in[8];
  const float* ad1    = (const float*)d_in[9];
  const float* b1     = (const float*)d_in[10];
  const float* lpW[3] = {(const float*)d_in[11], (const float*)d_in[13], (const float*)d_in[15]};
  const float* lpb[3] = {(const float*)d_in[12], (const float*)d_in[14], (const float*)d_in[16]};
  const float* kW[3]  = {(const float*)d_in[17], (const float*)d_in[19], (const float*)d_in[21]};
  const float* kb[3]  = {(const float*)d_in[18], (const float*)d_in[20], (const float*)d_in[22]};
  const float* pool_w = (const float*)d_in[23];
  const float* beta   = (const float*)d_in[24];
  const float* h0     = (const float*)d_in[25];

  float* ws    = (float*)d_ws;
  float* xh    = ws;                      // 262144
  float* cur1  = xh    + NODES * DIM;     // 262144
  float* cur2  = cur1  + NODES * DIM;     // 262144
  float* al_s  = cur2  + NODES * DIM;     // 8192
  float* al_d  = al_s  + NODES * HEADS;   // 8192
  float* e_max = al_d  + NODES * HEADS;   // 8192
  float* denom = e_max + NODES * HEADS;   // 8192
  float* ebuf  = denom + NODES * HEADS;   // 139264
  float* hr    = ebuf  + EPN * HEADS;     // 3*128
  float* kr    = hr    + 3 * NGRAPH * OUTD;
  float* mnb   = kr    + 3 * NGRAPH * OUTD;
  float* mxb   = mnb   + NGRAPH * DIM;
  float* hbd   = mxb   + NGRAPH * DIM;
  float* kf    = hbd   + NGRAPH * DIM;    // 10240

  const int* src = eidx;
  const int* dst = eidx + NEDGE;

  auto layer = [&](const float* cur, const float* W, const float* as,
                   const float* ad, const float* bias, float* outbuf, bool relu) {
    gemm_wmma_f32<<<NODES / 16, 256, 0, stream>>>(cur, W, xh);
    att_logits<<<(NODES * HEADS) / 256, 256, 0, stream>>>(xh, as, ad, al_s, al_d);
    init_layer<<<(NODES * DIM) / 256, 256, 0, stream>>>(bias, outbuf, e_max, denom);
    edge_max_k<<<(EPN * HEADS) / 256, 256, 0, stream>>>(src, dst, al_s, al_d, e_max);
    edge_exp_k<<<(EPN * HEADS) / 256, 256, 0, stream>>>(src, dst, al_s, al_d, e_max, ebuf, denom);
    edge_scatter_k<<<(EPN * DIM) / 256, 256, 0, stream>>>(src, dst, ebuf, denom, xh, outbuf);
    if (relu) relu_k<<<(NODES * DIM) / 256, 256, 0, stream>>>(outbuf);
  };

  auto readout = [&](const float* cur, int i) {
    readout_stats<<<NGRAPH, 128, 0, stream>>>(cur, pool_w, lpW[i], lpb[i],
                                              hr + i * NGRAPH * OUTD, mnb, mxb, hbd);
    kde_density<<<NGRAPH * DIM, 256, 0, stream>>>(cur, mnb, mxb, hbd, kf);
    kde_proj<<<NGRAPH, 32, 0, stream>>>(kf, kW[i], kb[i], kr + i * NGRAPH * OUTD);
  };

  readout(x, 0);
  layer(x, W0, as0, ad0, b0, cur1, true);
  readout(cur1, 1);
  layer(cur1, W1, as1, ad1, b1, cur2, false);
  readout(cur2, 2);
  final_risk<<<1, 32, 0, stream>>>(hr, kr, beta, h0, (float*)d_out);
}